// Affine_Linear_Abla_Eulur_44478681317764
// MI455X (gfx1250) — compile-verified
//
#include <hip/hip_runtime.h>

typedef __attribute__((ext_vector_type(2))) float v2f;
typedef __attribute__((ext_vector_type(8))) float v8f;

#define TOK_PER_GRP 16
#define GRP_PER_BLK 4
#define COLS 48        // 16 tokens * 3 output components
#define THREADS 384    // 12 waves = 4 M-tiles x 3 N-tiles

// Y(64 x 48cols) = A@Ta + Bw@Tb + C@Tc per 16-token group, via V_WMMA_F32_16X16X4_F32.
__global__ __launch_bounds__(THREADS)
void fused_euler_linear_kernel(const float* __restrict__ X,
                               const float* __restrict__ J,
                               const float* __restrict__ Wa,
                               const float* __restrict__ Wb,
                               const float* __restrict__ Wc,
                               float* __restrict__ Y)
{
    // Term matrices staged as K-pairs: T[term][e/2][col][e&1] -> B-frag = one b64 load
    __shared__ float T[3][32][COLS][2];   // 36 KB

    const int tid  = threadIdx.x;
    const int lane = tid & 31;
    const int wid  = tid >> 5;     // 0..11
    const int m    = wid & 3;      // f-tile   (rows m*16..m*16+15)
    const int n    = wid >> 2;     // col-tile (cols n*16..n*16+15)

    const float* wptr[3] = { Wa, Wb, Wc };

    // ---- Preload A-fragments (weights) into registers: 3 terms x 16 K-steps.
    // A 16x4 f32 layout (ISA 7.12.2): lanes 0-15 row=lane, vgpr0=K0,vgpr1=K1;
    //                                 lanes 16-31 row=lane-16, vgpr0=K2,vgpr1=K3.
    v2f afrag[3][16];
    {
        const int row = m * 16 + (lane & 15);
        const int ehi = (lane >> 4) * 2;
#pragma unroll
        for (int t = 0; t < 3; ++t) {
#pragma unroll
            for (int k = 0; k < 16; ++k) {
                afrag[t][k] = *(const v2f*)(wptr[t] + row * 64 + k * 4 + ehi);
            }
        }
    }

    for (int gi = 0; gi < GRP_PER_BLK; ++gi) {
        const long grp     = (long)blockIdx.x * GRP_PER_BLK + gi;
        const long tokBase = grp * TOK_PER_GRP;

        // ---- Phase 1: per-(token,e) rotation terms -> LDS
        for (int base = 0; base < TOK_PER_GRP * 64; base += THREADS) {
            const int idx = base + tid;
            if (idx < TOK_PER_GRP * 64) {
                const int  tl  = idx >> 6;
                const int  e   = idx & 63;
                const long tok = tokBase + tl;
                const float* Xp = X + tok * 192 + e * 3;
                const float* Jp = J + tok * 192 + e * 3;
                if (gi + 1 < GRP_PER_BLK) {      // pull next group's lines into cache
                    __builtin_prefetch(Xp + TOK_PER_GRP * 192, 0, 1);
                    __builtin_prefetch(Jp + TOK_PER_GRP * 192, 0, 1);
                }
                const float x0 = Xp[0], x1 = Xp[1], x2 = Xp[2];
                float sa, ca, sb, cb, sg, cg;
                __sincosf(Jp[0], &sa, &ca);
                __sincosf(Jp[1], &sb, &cb);
                __sincosf(Jp[2], &sg, &cg);
                const float r00 = ca * cb;
                const float r01 = ca * sb * sg - sa * cg;
                const float r02 = ca * sb * cg + sa * sg;
                const float r10 = sa * cb;
                const float r11 = sa * sb * sg + ca * cg;
                const float r12 = sa * sb * cg - ca * sg;
                const float r20 = -sb;
                const float r21 = cb * sg;
                const float r22 = cb * cg;
                // v_i = sum_k R[k][i] * x[k]
                const float v0 = r00 * x0 + r10 * x1 + r20 * x2;
                const float v1 = r01 * x0 + r11 * x1 + r21 * x2;
                const float v2 = r02 * x0 + r12 * x1 + r22 * x2;
                // a = col0*v0 + col1*v1 ; b = col1*v0 - col0*v1 ; c = col2*v2
                const float a0 = r00 * v0 + r01 * v1;
                const float a1 = r10 * v0 + r11 * v1;
                const float a2 = r20 * v0 + r21 * v1;
                const float b0 = r01 * v0 - r00 * v1;
                const float b1 = r11 * v0 - r10 * v1;
                const float b2 = r21 * v0 - r20 * v1;
                const float c0 = r02 * v2;
                const float c1 = r12 * v2;
                const float c2 = r22 * v2;
                const int k2 = e >> 1, lo = e & 1, c = tl * 3;
                T[0][k2][c + 0][lo] = a0; T[0][k2][c + 1][lo] = a1; T[0][k2][c + 2][lo] = a2;
                T[1][k2][c + 0][lo] = b0; T[1][k2][c + 1][lo] = b1; T[1][k2][c + 2][lo] = b2;
                T[2][k2][c + 0][lo] = c0; T[2][k2][c + 1][lo] = c1; T[2][k2][c + 2][lo] = c2;
            }
        }
        __syncthreads();

        // ---- Phase 2: three independent WMMA accumulation chains (depth 16 each)
        v8f acc3[3];
        {
            // B 4x16 f32 layout: lanes 0-15 get (K0,K1), lanes 16-31 get (K2,K3)
            const int bc = n * 16 + (lane & 15);
            const int bk = lane >> 4;
#pragma unroll
            for (int t = 0; t < 3; ++t) {
                acc3[t] = (v8f){0.f, 0.f, 0.f, 0.f, 0.f, 0.f, 0.f, 0.f};
#pragma unroll
                for (int k = 0; k < 16; ++k) {
                    const v2f b = *(const v2f*)&T[t][k * 2 + bk][bc][0];
                    acc3[t] = __builtin_amdgcn_wmma_f32_16x16x4_f32(
                        false, afrag[t][k], false, b, (short)0, acc3[t], false, false);
                }
            }
        }
        const v8f acc = acc3[0] + acc3[1] + acc3[2];

        // ---- Store: D vgpr r -> row (lane<16 ? r : r+8), col = lane%16
        {
            const int c     = n * 16 + (lane & 15);
            const int tl    = c / 3;
            const int i     = c % 3;
            const int fbase = m * 16 + (lane >> 4) * 8;
            float* outp = Y + (tokBase + tl) * 192 + i;
#pragma unroll
            for (int r = 0; r < 8; ++r)
                outp[(fbase + r) * 3] = acc[r];
        }
        __syncthreads();
    }
}

extern "C" void kernel_launch(void* const* d_in, const int* in_sizes, int n_in,
                              void* d_out, int out_size, void* d_ws, size_t ws_size,
                              hipStream_t stream) {
    (void)in_sizes; (void)n_in; (void)d_ws; (void)ws_size; (void)out_size;
    const float* X  = (const float*)d_in[0];
    const float* J  = (const float*)d_in[1];
    const float* Wa = (const float*)d_in[2];
    const float* Wb = (const float*)d_in[3];
    const float* Wc = (const float*)d_in[4];
    float* Y = (float*)d_out;
    // B*N = 131072 tokens = 2048 blocks * 4 groups * 16 tokens
    fused_euler_linear_kernel<<<2048, THREADS, 0, stream>>>(X, J, Wa, Wb, Wc, Y);
}